// QuantumLayer_53317724013128
// MI455X (gfx1250) — compile-verified
//
#include <hip/hip_runtime.h>
#include <math.h>

typedef __attribute__((ext_vector_type(2))) float v2f;
typedef __attribute__((ext_vector_type(8))) float v8f;

#define N_QUBITS 16
#define D_IN 128
#define D_OUT 128

// S_q = { i <= q : (q - i) % 4 in {0,1} }  (rows of L^3 over GF(2), L = prefix-xor)
__device__ __forceinline__ bool in_mask(int q, int i) {
    return (i <= q) && (((q - i) & 3) < 2);
}

__global__ __launch_bounds__(256) void quantum_layer_fused(
    const float* __restrict__ x,      // (B, 128)
    const float* __restrict__ W_in,   // (16, 128)
    const float* __restrict__ b_in,   // (16)
    const float* __restrict__ W_out,  // (128, 16)
    const float* __restrict__ b_out,  // (128)
    float* __restrict__ out)          // (B, 128)
{
    __shared__ float cth[16][17];   // cos(theta) tile, padded
    __shared__ float expv[16][17];  // expectation values tile, padded

    const int tid   = threadIdx.x;
    const int wave  = tid >> 5;     // 0..7
    const int lane  = tid & 31;
    const int l15   = lane & 15;
    const int lhalf = lane >> 4;    // 0 or 1 (selects K pair in f32 WMMA A/B layout)
    const int row0  = blockIdx.x * 16;

    // ---- Phase 1 (wave 0): theta tile = x[row0:row0+16] @ W_in^T, then cos(theta+b_in) -> LDS
    if (wave == 0) {
        v8f acc = {};
        // A: 16x4 f32 tile of x.  lane<16: M=lane, K=(kb,kb+1); lane>=16: M=lane-16, K=(kb+2,kb+3)
        const float* xrow = x    + (size_t)(row0 + l15) * D_IN + 2 * lhalf;
        // B: 4x16 f32 tile of W_in^T: B[k][n] = W_in[n][k]
        const float* wrow = W_in + (size_t)l15 * D_IN + 2 * lhalf;
        #pragma unroll
        for (int kb = 0; kb < D_IN; kb += 4) {
            v2f a = *(const v2f*)(xrow + kb);
            v2f b = *(const v2f*)(wrow + kb);
            acc = __builtin_amdgcn_wmma_f32_16x16x4_f32(
                false, a, false, b, (short)0, acc, false, false);
        }
        // C layout: VGPR r, lane<16 -> (M=r, N=lane); lane>=16 -> (M=r+8, N=lane-16)
        const float bq = b_in[l15];
        #pragma unroll
        for (int r = 0; r < 8; ++r) {
            const int m = r + 8 * lhalf;
            cth[m][l15] = cosf(acc[r] + bq);
        }
    }
    __syncthreads();

    // ---- Phase 2 (all 256 threads): expv[m][q] = prod_{i in S_q} cos(theta[m][i])
    {
        const int m = tid >> 4;   // 0..15
        const int q = tid & 15;   // 0..15
        float p = 1.0f;
        #pragma unroll
        for (int i = 0; i < N_QUBITS; ++i) {
            const float f = in_mask(q, i) ? cth[m][i] : 1.0f;
            p *= f;
        }
        expv[m][q] = p;
    }
    __syncthreads();

    // ---- Phase 3 (each wave = one 16-wide N tile): out = expv @ W_out^T + b_out
    {
        const int nt = wave;                 // N tile 0..7 -> columns nt*16 .. nt*16+15
        const int n  = nt * 16 + l15;
        v8f acc = {};
        // B: 4x16 tile of W_out^T: B[k][n] = W_out[n][k]
        const float* wo = W_out + (size_t)n * N_QUBITS + 2 * lhalf;
        #pragma unroll
        for (int kc = 0; kc < N_QUBITS; kc += 4) {
            v2f a;
            a.x = expv[l15][kc + 2 * lhalf];
            a.y = expv[l15][kc + 2 * lhalf + 1];
            v2f b = *(const v2f*)(wo + kc);
            acc = __builtin_amdgcn_wmma_f32_16x16x4_f32(
                false, a, false, b, (short)0, acc, false, false);
        }
        const float bo = b_out[n];
        #pragma unroll
        for (int r = 0; r < 8; ++r) {
            const int m = r + 8 * lhalf;
            out[(size_t)(row0 + m) * D_OUT + n] = acc[r] + bo;
        }
    }
}

extern "C" void kernel_launch(void* const* d_in, const int* in_sizes, int n_in,
                              void* d_out, int out_size, void* d_ws, size_t ws_size,
                              hipStream_t stream) {
    const float* x     = (const float*)d_in[0];  // (B,128)
    const float* W_in  = (const float*)d_in[1];  // (16,128)
    const float* b_in  = (const float*)d_in[2];  // (16)
    // d_in[3] = q_weights: provably irrelevant to the forward output
    // (diagonal phases have unit modulus; CNOT chains only permute probabilities).
    const float* W_out = (const float*)d_in[4];  // (128,16)
    const float* b_out = (const float*)d_in[5];  // (128)
    float* out = (float*)d_out;

    const int B = in_sizes[0] / D_IN;            // 256
    dim3 grid(B / 16), block(256);
    hipLaunchKernelGGL(quantum_layer_fused, grid, block, 0, stream,
                       x, W_in, b_in, W_out, b_out, out);
}